// DCGNN_3487513444705
// MI455X (gfx1250) — compile-verified
//
#include <hip/hip_runtime.h>

typedef float v2f __attribute__((ext_vector_type(2)));
typedef float v8f __attribute__((ext_vector_type(8)));

#define NB    2048
#define C_    62
#define FIN   512
#define FOUT  256
#define CP    64
#define KCH   5
#define THRESH_ 0.1f

// ws layout (floats)
#define WS_ARAW   0          // 62*62  (padded region 4096)
#define WS_LT     4096       // 64*64 L^T padded
#define WS_PHI    8192       // 4 planes * 64*512  (Phi_1..Phi_4)
#define WS_H      139264     // 2 planes * 62*512 = 63488
#define WS_DCONST 202752     // 2 floats

// ---------------------------------------------------------------------------
// Kernel 1: learned adjacency  W* = mlp(coord); A = (W* > t) & ~eye ? W* : 0
// ---------------------------------------------------------------------------
__global__ void adj_kernel(const float* __restrict__ coord,
                           const float* __restrict__ w1, const float* __restrict__ b1,
                           const float* __restrict__ w2, const float* __restrict__ b2,
                           float* __restrict__ Araw) {
  int idx = blockIdx.x * blockDim.x + threadIdx.x;
  if (idx >= C_ * C_) return;
  const float c0 = coord[idx * 4 + 0], c1 = coord[idx * 4 + 1];
  const float c2 = coord[idx * 4 + 2], c3 = coord[idx * 4 + 3];
  float acc = b2[0];
#pragma unroll 4
  for (int m = 0; m < 64; ++m) {
    float h = b1[m] + c0 * w1[m] + c1 * w1[64 + m] + c2 * w1[128 + m] + c3 * w1[192 + m];
    h = fmaxf(h, 0.f);
    acc += h * w2[m];
  }
  const int i = idx / C_, j = idx - i * C_;
  Araw[idx] = (acc > THRESH_ && i != j) ? acc : 0.f;
}

// ---------------------------------------------------------------------------
// Kernel 2: L^T = (-D^-1/2 A D^-1/2)^T, zero-padded to 64x64
// ---------------------------------------------------------------------------
__global__ void lapT_kernel(const float* __restrict__ Araw, float* __restrict__ LTp) {
  __shared__ float dis[C_];
  const int tid = threadIdx.x;
  if (tid < C_) {
    float dg = 0.f;
    for (int j = 0; j < C_; ++j) dg += Araw[tid * C_ + j];
    dis[tid] = dg > 0.f ? rsqrtf(dg) : 0.f;
  }
  __syncthreads();
  for (int idx = tid; idx < CP * CP; idx += blockDim.x) {
    const int r = idx >> 6, cc = idx & 63;       // LT[r,cc] = L[cc,r]
    float v = 0.f;
    if (r < C_ && cc < C_) v = -(dis[cc] * Araw[cc * C_ + r] * dis[r]);
    LTp[idx] = v;
  }
}

// ---------------------------------------------------------------------------
// Kernel 3: Phi_k = T_k(L^T) applied to F = fc_w reshaped [62,512]
//           (columns independent -> 8 blocks, 64-column slices, f32 WMMA)
// ---------------------------------------------------------------------------
__global__ void __launch_bounds__(128)
phi_kernel(const float* __restrict__ fcw, const float* __restrict__ LTp,
           float* __restrict__ phi) {
  __shared__ float Pa[CP * CP], Pb[CP * CP], Lsh[CP * CP];
  const int tid = threadIdx.x;
  const int w = tid >> 5, lane = tid & 31, lh = lane >> 4, lm = lane & 15;
  const int s = blockIdx.x;                      // column slice [64s, 64s+64)
  for (int i = tid; i < CP * CP; i += 128) {
    const int c = i >> 6, t = i & 63;
    Pa[i]  = (c < C_) ? fcw[c * 512 + s * 64 + t] : 0.f;
    Lsh[i] = LTp[i];
  }
  __syncthreads();

  const v8f vz = {0.f, 0.f, 0.f, 0.f, 0.f, 0.f, 0.f, 0.f};

  auto lap64 = [&](float* Dst, const float* Src, bool set) {
    const int n0 = w * 16 + lm;                  // wave owns one 16-col strip
    v8f d[4];
#pragma unroll
    for (int mt = 0; mt < 4; ++mt) d[mt] = vz;
#pragma unroll 4
    for (int kk = 0; kk < CP; kk += 4) {
      v2f bf;
      bf.x = Src[(kk + 2 * lh + 0) * CP + n0];
      bf.y = Src[(kk + 2 * lh + 1) * CP + n0];
#pragma unroll
      for (int mt = 0; mt < 4; ++mt) {
        v2f a = *(const v2f*)&Lsh[(mt * 16 + lm) * CP + kk + 2 * lh];
        d[mt] = __builtin_amdgcn_wmma_f32_16x16x4_f32(
            false, a, false, bf, (short)0, d[mt], false, false);
      }
    }
#pragma unroll
    for (int mt = 0; mt < 4; ++mt)
#pragma unroll
      for (int r = 0; r < 8; ++r) {
        const int c = mt * 16 + r + 8 * lh;
        const int idx = c * CP + n0;
        const float p = d[mt][r];
        Dst[idx] = set ? p : (2.f * p - Dst[idx]);
      }
  };
  auto dump = [&](const float* Src, int k) {
    for (int i = tid; i < CP * CP; i += 128) {
      const int c = i >> 6, t = i & 63;
      phi[(k - 1) * (CP * 512) + c * 512 + s * 64 + t] = Src[i];
    }
  };

  lap64(Pb, Pa, true);  __syncthreads(); dump(Pb, 1);   // Phi_1
  lap64(Pa, Pb, false); __syncthreads(); dump(Pa, 2);   // Phi_2
  lap64(Pb, Pa, false); __syncthreads(); dump(Pb, 3);   // Phi_3
  lap64(Pa, Pb, false); __syncthreads(); dump(Pa, 4);   // Phi_4
}

// ---------------------------------------------------------------------------
// Kernel 4: H[:, :, n] = sum_k Phi_k_n @ W_k^T   (one 16x16 WMMA tile / wave)
//   256 waves: g -> n = g&1, mt = (g>>1)&3, nt = g>>3
// ---------------------------------------------------------------------------
__global__ void __launch_bounds__(256)
hacc_kernel(const float* __restrict__ fcw, const float* __restrict__ phi,
            const float* __restrict__ chW, float* __restrict__ Hws) {
  const int tid = threadIdx.x;
  const int w = tid >> 5, lane = tid & 31, lh = lane >> 4, lm = lane & 15;
  const int g = blockIdx.x * 8 + w;
  const int n = g & 1, mt = (g >> 1) & 3, nt = g >> 3;

  const int jrow = mt * 16 + lm;
  const int arow = (jrow < C_ ? jrow : C_ - 1) * 512;  // clamp: rows 62/63 discarded
  const v8f vz = {0.f, 0.f, 0.f, 0.f, 0.f, 0.f, 0.f, 0.f};
  v8f d = vz;

  for (int k = 0; k < KCH; ++k) {
    const float* As = (k == 0) ? fcw : (phi + (k - 1) * (CP * 512));
    const float* Bs = chW + k * (FIN * FOUT);
#pragma unroll 4
    for (int kk = 0; kk < FOUT; kk += 4) {
      v2f a, bf;
      a.x = As[arow + (kk + 2 * lh) * 2 + n];
      a.y = As[arow + (kk + 2 * lh) * 2 + n + 2];
      bf  = *(const v2f*)&Bs[(nt * 16 + lm) * FOUT + kk + 2 * lh];
      d = __builtin_amdgcn_wmma_f32_16x16x4_f32(
          false, a, false, bf, (short)0, d, false, false);
    }
  }
#pragma unroll
  for (int r = 0; r < 8; ++r) {
    const int j = mt * 16 + r + 8 * lh;
    if (j < C_) Hws[n * (C_ * 512) + j * 512 + nt * 16 + lm] = d[r];
  }
}

// ---------------------------------------------------------------------------
// Kernel 5: dconst[n] = cw * <cheb_b, F_n> + cb * sum(F_n) + fc_b[n]
// ---------------------------------------------------------------------------
__global__ void dconst_kernel(const float* __restrict__ chebb, const float* __restrict__ fcw,
                              const float* __restrict__ cvw, const float* __restrict__ cvb,
                              const float* __restrict__ fcb, float* __restrict__ dws) {
  __shared__ float red[32];
  const int tid = threadIdx.x, w = tid >> 5, lane = tid & 31;
  float s0 = 0.f, s1 = 0.f, t0 = 0.f, t1 = 0.f;
  for (int cf = tid; cf < C_ * FOUT; cf += 256) {
    const int f = cf & 255;
    const float bb = chebb[f];
    const float w0 = fcw[2 * cf], w1 = fcw[2 * cf + 1];
    s0 += bb * w0; s1 += bb * w1; t0 += w0; t1 += w1;
  }
#pragma unroll
  for (int off = 16; off > 0; off >>= 1) {
    s0 += __shfl_down(s0, off, 32); s1 += __shfl_down(s1, off, 32);
    t0 += __shfl_down(t0, off, 32); t1 += __shfl_down(t1, off, 32);
  }
  if (lane == 0) { red[w * 4] = s0; red[w * 4 + 1] = s1; red[w * 4 + 2] = t0; red[w * 4 + 3] = t1; }
  __syncthreads();
  if (tid == 0) {
    float S0 = 0, S1 = 0, T0 = 0, T1 = 0;
    for (int i = 0; i < 8; ++i) {
      S0 += red[i * 4]; S1 += red[i * 4 + 1]; T0 += red[i * 4 + 2]; T1 += red[i * 4 + 3];
    }
    dws[0] = cvw[0] * S0 + cvb[0] * T0 + fcb[0];
    dws[1] = cvw[0] * S1 + cvb[0] * T1 + fcb[1];
  }
}

// ---------------------------------------------------------------------------
// Kernel 6: the memory-bound pass: out[b,n] = cw * <x[b], H_n> + dconst[n]
//   4 batch rows / block, b128 streaming + prefetch, deterministic reduce
// ---------------------------------------------------------------------------
__global__ void __launch_bounds__(256)
proj_kernel(const float* __restrict__ x, const float* __restrict__ Hws,
            const float* __restrict__ cvw, const float* __restrict__ dws,
            float* __restrict__ out) {
  __shared__ float red[64];
  const int tid = threadIdx.x, w = tid >> 5, lane = tid & 31;
  const int b0 = blockIdx.x * 4;

  const float4* H0v = (const float4*)Hws;                       // 7936 vec4
  const float4* H1v = (const float4*)(Hws + C_ * 512);
  const float4* xr0 = (const float4*)(x + (size_t)(b0 + 0) * (C_ * FIN));
  const float4* xr1 = (const float4*)(x + (size_t)(b0 + 1) * (C_ * FIN));
  const float4* xr2 = (const float4*)(x + (size_t)(b0 + 2) * (C_ * FIN));
  const float4* xr3 = (const float4*)(x + (size_t)(b0 + 3) * (C_ * FIN));

  float p0[4] = {0.f, 0.f, 0.f, 0.f};
  float p1[4] = {0.f, 0.f, 0.f, 0.f};

  for (int i = tid; i < (C_ * FIN) / 4; i += 256) {
    __builtin_prefetch(xr0 + i + 512, 0, 0);
    __builtin_prefetch(xr1 + i + 512, 0, 0);
    __builtin_prefetch(xr2 + i + 512, 0, 0);
    __builtin_prefetch(xr3 + i + 512, 0, 0);
    const float4 h0 = H0v[i], h1 = H1v[i];
    const float4 v0 = xr0[i], v1 = xr1[i], v2 = xr2[i], v3 = xr3[i];
    p0[0] += v0.x * h0.x + v0.y * h0.y + v0.z * h0.z + v0.w * h0.w;
    p1[0] += v0.x * h1.x + v0.y * h1.y + v0.z * h1.z + v0.w * h1.w;
    p0[1] += v1.x * h0.x + v1.y * h0.y + v1.z * h0.z + v1.w * h0.w;
    p1[1] += v1.x * h1.x + v1.y * h1.y + v1.z * h1.z + v1.w * h1.w;
    p0[2] += v2.x * h0.x + v2.y * h0.y + v2.z * h0.z + v2.w * h0.w;
    p1[2] += v2.x * h1.x + v2.y * h1.y + v2.z * h1.z + v2.w * h1.w;
    p0[3] += v3.x * h0.x + v3.y * h0.y + v3.z * h0.z + v3.w * h0.w;
    p1[3] += v3.x * h1.x + v3.y * h1.y + v3.z * h1.z + v3.w * h1.w;
  }
#pragma unroll
  for (int r = 0; r < 4; ++r)
#pragma unroll
    for (int off = 16; off > 0; off >>= 1) {
      p0[r] += __shfl_down(p0[r], off, 32);
      p1[r] += __shfl_down(p1[r], off, 32);
    }
  if (lane == 0) {
#pragma unroll
    for (int r = 0; r < 4; ++r) {
      red[w * 8 + 2 * r]     = p0[r];
      red[w * 8 + 2 * r + 1] = p1[r];
    }
  }
  __syncthreads();
  if (tid < 8) {
    const int r = tid >> 1, n = tid & 1;
    float s = 0.f;
#pragma unroll
    for (int ww = 0; ww < 8; ++ww) s += red[ww * 8 + 2 * r + n];
    out[(b0 + r) * 2 + n] = cvw[0] * s + dws[n];
  }
}

// ---------------------------------------------------------------------------
extern "C" void kernel_launch(void* const* d_in, const int* in_sizes, int n_in,
                              void* d_out, int out_size, void* d_ws, size_t ws_size,
                              hipStream_t stream) {
  const float* x     = (const float*)d_in[0];
  const float* coord = (const float*)d_in[1];
  const float* aw1   = (const float*)d_in[2];
  const float* ab1   = (const float*)d_in[3];
  const float* aw2   = (const float*)d_in[4];
  const float* ab2   = (const float*)d_in[5];
  const float* chW   = (const float*)d_in[6];
  const float* chb   = (const float*)d_in[7];
  const float* cvw   = (const float*)d_in[8];
  const float* cvb   = (const float*)d_in[9];
  const float* fcw   = (const float*)d_in[10];
  const float* fcb   = (const float*)d_in[11];
  float* out = (float*)d_out;

  float* ws    = (float*)d_ws;
  float* Araw  = ws + WS_ARAW;
  float* LTp   = ws + WS_LT;
  float* phi   = ws + WS_PHI;
  float* Hws   = ws + WS_H;
  float* dws   = ws + WS_DCONST;

  adj_kernel<<<(C_ * C_ + 127) / 128, 128, 0, stream>>>(coord, aw1, ab1, aw2, ab2, Araw);
  lapT_kernel<<<1, 256, 0, stream>>>(Araw, LTp);
  phi_kernel<<<8, 128, 0, stream>>>(fcw, LTp, phi);
  hacc_kernel<<<32, 256, 0, stream>>>(fcw, phi, chW, Hws);
  dconst_kernel<<<1, 256, 0, stream>>>(chb, fcw, cvw, cvb, fcb, dws);
  proj_kernel<<<NB / 4, 256, 0, stream>>>(x, Hws, cvw, dws, out);
}